// MambaBlock_74337293959197
// MI455X (gfx1250) — compile-verified
//
#include <hip/hip_runtime.h>
#include <hip/hip_bf16.h>
#include <math.h>

// ---------------------------------------------------------------------------
// MambaBlock on gfx1250 (MI455X).  fp32 WMMA (V_WMMA_F32_16X16X4_F32) for all
// five GEMMs with 32x64 register-blocked macro-tiles per wave (8 WMMAs per
// k-step sharing A/B fragments); scan parallelized over B*DI*DS lanes with
// 16-lane shuffle reduction; SiLU gate fused into the scan epilogue.
// ---------------------------------------------------------------------------

constexpr int CB  = 4;
constexpr int CD  = 512;
constexpr int CL  = 2048;
constexpr int CDI = 1024;   // d_inner
constexpr int CDS = 16;     // d_state
constexpr int CK  = 7;      // conv kernel
constexpr int CDT = 64;     // dt_rank

typedef __attribute__((ext_vector_type(2))) float v2f;
typedef __attribute__((ext_vector_type(8))) float v8f;

// ---------------------------------------------------------------------------
// MT x NT register-blocked WMMA macro-tile:
//   acc[mi][nj][m,n] = sum_k A[mi*16+m, k] * W[nj*16+n, k]
// A element (m,k) at A[m*sM + k*sK];  W row-major (N,K), offset to n0, ld ldW.
// fp32 16x16x4 fragment layout (ISA 7.12.2): lanes 0-15 hold k=(0,1),
// lanes 16-31 hold k=(2,3), row/col = lane&15.
// C/D: lane = column n, VGPR i = row (i + 8*(lane>>4)).
// ---------------------------------------------------------------------------
template <int MT, int NT>
__device__ __forceinline__ void wmma_macro_tile(const float* __restrict__ A,
                                                long sM, long sK,
                                                const float* __restrict__ Wt,
                                                int ldW, int K,
                                                v8f acc[MT][NT]) {
  const int lane = threadIdx.x & 31;
  const int half = lane >> 4;
  const int r    = lane & 15;

  const float* Ar[MT];
  const float* Wr[NT];
#pragma unroll
  for (int mi = 0; mi < MT; ++mi)
    Ar[mi] = A + (long)(mi * 16 + r) * sM + (long)(2 * half) * sK;
#pragma unroll
  for (int nj = 0; nj < NT; ++nj)
    Wr[nj] = Wt + (long)(nj * 16 + r) * ldW + (2 * half);

  for (int k0 = 0; k0 < K; k0 += 4) {
    if ((k0 & 63) == 0 && (k0 + 64) < K) {
#pragma unroll
      for (int mi = 0; mi < MT; ++mi)
        __builtin_prefetch(Ar[mi] + (long)(k0 + 64) * sK, 0, 3);
#pragma unroll
      for (int nj = 0; nj < NT; ++nj)
        __builtin_prefetch(Wr[nj] + (k0 + 64), 0, 3);
    }
    v2f a[MT], b[NT];
#pragma unroll
    for (int mi = 0; mi < MT; ++mi) {
      a[mi].x = Ar[mi][(long)k0 * sK];
      a[mi].y = Ar[mi][(long)(k0 + 1) * sK];
    }
#pragma unroll
    for (int nj = 0; nj < NT; ++nj) {
      b[nj].x = Wr[nj][k0];
      b[nj].y = Wr[nj][k0 + 1];
    }
#pragma unroll
    for (int mi = 0; mi < MT; ++mi)
#pragma unroll
      for (int nj = 0; nj < NT; ++nj)
        acc[mi][nj] = __builtin_amdgcn_wmma_f32_16x16x4_f32(
            false, a[mi], false, b[nj], (short)0, acc[mi][nj], false, false);
  }
}

__device__ __forceinline__ float silu_f(float t) {
  return t * (1.0f / (1.0f + expf(-t)));
}

// ---------------------------------------------------------------------------
// 1) in-proj:  proj[b,l,e] = sum_d x[b,d,l]*W_in[e,d] + b_in[e]
//    e <  DI -> a_pre (B,DI,L);  e >= DI -> bg (B,L,DI)
// ---------------------------------------------------------------------------
__global__ __launch_bounds__(128) void k_in_proj(const float* __restrict__ x,
                                                 const float* __restrict__ W_in,
                                                 const float* __restrict__ b_in,
                                                 float* __restrict__ a_pre,
                                                 float* __restrict__ bg) {
  const int l0 = (blockIdx.x * 4 + threadIdx.y) * 32;
  const int n0 = blockIdx.y * 64;
  const int b  = blockIdx.z;
  const float* A = x + (size_t)b * CD * CL + l0;        // A[l,d] = x[b,d,l0+l]
  v8f acc[2][4] = {};
  wmma_macro_tile<2, 4>(A, 1, CL, W_in + (size_t)n0 * CD, CD, CD, acc);
  const int lane = threadIdx.x & 31, half = lane >> 4, r = lane & 15;
#pragma unroll
  for (int nj = 0; nj < 4; ++nj) {
    const int e = n0 + nj * 16 + r;
    const float bias = b_in[e];
#pragma unroll
    for (int mi = 0; mi < 2; ++mi)
#pragma unroll
      for (int i = 0; i < 8; ++i) {
        const int l = l0 + mi * 16 + 8 * half + i;
        const float v = acc[mi][nj][i] + bias;
        if (e < CDI) a_pre[((size_t)b * CDI + e) * CL + l] = v;
        else         bg[((size_t)b * CL + l) * CDI + (e - CDI)] = v;
      }
  }
}

// ---------------------------------------------------------------------------
// 2) depthwise conv (K=7, same padding) over L, per channel
// ---------------------------------------------------------------------------
__global__ __launch_bounds__(256) void k_dwconv(const float* __restrict__ a_pre,
                                                const float* __restrict__ dw_w,
                                                const float* __restrict__ dw_b,
                                                float* __restrict__ a_conv) {
  const size_t idx = (size_t)blockIdx.x * blockDim.x + threadIdx.x;
  if (idx >= (size_t)CB * CDI * CL) return;
  const int l = (int)(idx % CL);
  const int d = (int)((idx / CL) % CDI);
  const size_t ch = idx / CL;            // b*CDI + d
  float s = dw_b[d];
#pragma unroll
  for (int j = 0; j < CK; ++j) {
    const int ll = l + j - CK / 2;
    if (ll >= 0 && ll < CL) s += a_pre[ch * CL + ll] * dw_w[d * CK + j];
  }
  a_conv[idx] = s;
}

// ---------------------------------------------------------------------------
// 3) pointwise conv + SiLU:  u[b,l,e] = silu(sum_i a_conv[b,i,l]*pw_w[e,i] + pw_b[e])
// ---------------------------------------------------------------------------
__global__ __launch_bounds__(128) void k_pw_silu(const float* __restrict__ a_conv,
                                                 const float* __restrict__ pw_w,
                                                 const float* __restrict__ pw_b,
                                                 float* __restrict__ u) {
  const int l0 = (blockIdx.x * 4 + threadIdx.y) * 32;
  const int n0 = blockIdx.y * 64;
  const int b  = blockIdx.z;
  const float* A = a_conv + (size_t)b * CDI * CL + l0;  // A[l,i] = a_conv[b,i,l0+l]
  v8f acc[2][4] = {};
  wmma_macro_tile<2, 4>(A, 1, CL, pw_w + (size_t)n0 * CDI, CDI, CDI, acc);
  const int lane = threadIdx.x & 31, half = lane >> 4, r = lane & 15;
#pragma unroll
  for (int nj = 0; nj < 4; ++nj) {
    const int e = n0 + nj * 16 + r;
    const float bias = pw_b[e];
#pragma unroll
    for (int mi = 0; mi < 2; ++mi)
#pragma unroll
      for (int i = 0; i < 8; ++i) {
        const int l = l0 + mi * 16 + 8 * half + i;
        u[((size_t)b * CL + l) * CDI + e] = silu_f(acc[mi][nj][i] + bias);
      }
  }
}

// ---------------------------------------------------------------------------
// 4) x-proj (N=96):  xdbl[b,l,e] = sum_d u[b,l,d]*W_x[e,d];  split dt / B / C
// ---------------------------------------------------------------------------
__global__ __launch_bounds__(128) void k_xproj(const float* __restrict__ u,
                                               const float* __restrict__ W_x,
                                               float* __restrict__ dtb,
                                               float* __restrict__ Bm,
                                               float* __restrict__ Cm) {
  const int l0 = (blockIdx.x * 4 + threadIdx.y) * 32;
  const int n0 = blockIdx.y * 32;
  const int b  = blockIdx.z;
  const float* A = u + ((size_t)b * CL + l0) * CDI;     // row-major (L, DI)
  v8f acc[2][2] = {};
  wmma_macro_tile<2, 2>(A, CDI, 1, W_x + (size_t)n0 * CDI, CDI, CDI, acc);
  const int lane = threadIdx.x & 31, half = lane >> 4, r = lane & 15;
#pragma unroll
  for (int nj = 0; nj < 2; ++nj) {
    const int e = n0 + nj * 16 + r;
#pragma unroll
    for (int mi = 0; mi < 2; ++mi)
#pragma unroll
      for (int i = 0; i < 8; ++i) {
        const int l = l0 + mi * 16 + 8 * half + i;
        const size_t base = (size_t)b * CL + l;
        const float v = acc[mi][nj][i];
        if (e < CDT)            dtb[base * CDT + e] = v;
        else if (e < CDT + CDS) Bm[base * CDS + (e - CDT)] = v;
        else                    Cm[base * CDS + (e - CDT - CDS)] = v;
      }
  }
}

// ---------------------------------------------------------------------------
// 5) dt-proj + softplus: delta[b,l,e] = softplus(sum_r dt[b,l,r]*W_dt[e,r] + b_dt[e])
// ---------------------------------------------------------------------------
__global__ __launch_bounds__(128) void k_dtproj(const float* __restrict__ dtb,
                                                const float* __restrict__ W_dt,
                                                const float* __restrict__ b_dt,
                                                float* __restrict__ delta) {
  const int l0 = (blockIdx.x * 4 + threadIdx.y) * 32;
  const int n0 = blockIdx.y * 64;
  const int b  = blockIdx.z;
  const float* A = dtb + ((size_t)b * CL + l0) * CDT;   // row-major (L, DT)
  v8f acc[2][4] = {};
  wmma_macro_tile<2, 4>(A, CDT, 1, W_dt + (size_t)n0 * CDT, CDT, CDT, acc);
  const int lane = threadIdx.x & 31, half = lane >> 4, r = lane & 15;
#pragma unroll
  for (int nj = 0; nj < 4; ++nj) {
    const int e = n0 + nj * 16 + r;
    const float bias = b_dt[e];
#pragma unroll
    for (int mi = 0; mi < 2; ++mi)
#pragma unroll
      for (int i = 0; i < 8; ++i) {
        const int l = l0 + mi * 16 + 8 * half + i;
        const float t = acc[mi][nj][i] + bias;
        const float sp = (t > 20.0f) ? t : log1pf(expf(t));
        delta[((size_t)b * CL + l) * CDI + e] = sp;
      }
  }
}

// ---------------------------------------------------------------------------
// 6) selective scan + Dskip + SiLU gate.
//    thread = (b, d, s); s = lane & 15; y reduced over the 16 states via
//    __shfl_xor(width=16);  g[b,l,d] = (y + Dskip[d]*u) * silu(bg[b,l,d])
//    (written over bg in-place: each element read/written by its own group)
// ---------------------------------------------------------------------------
__global__ __launch_bounds__(256) void k_scan_gate(const float* __restrict__ delta,
                                                   const float* __restrict__ Bm,
                                                   const float* __restrict__ Cm,
                                                   const float* __restrict__ u,
                                                   const float* __restrict__ A_log,
                                                   const float* __restrict__ Dskip,
                                                   float* __restrict__ g /* == bg */) {
  const int t = blockIdx.x * blockDim.x + threadIdx.x;   // B*DI*DS threads
  const int s = t & (CDS - 1);
  const int d = (t >> 4) & (CDI - 1);
  const int b = t >> 14;
  const float As = -expf(A_log[d * CDS + s]);
  const float Dd = Dskip[d];
  float h = 0.0f;
  for (int l = 0; l < CL; ++l) {
    const size_t base = (size_t)b * CL + l;
    const float dt_ = delta[base * CDI + d];   // broadcast within 16-lane group
    const float ut  = u[base * CDI + d];
    const float Bt  = Bm[base * CDS + s];
    const float Ct  = Cm[base * CDS + s];
    const float dA  = expf(dt_ * As);
    h = dA * h + (dt_ * ut) * Bt;
    float p = h * Ct;
    p += __shfl_xor(p, 1, 16);
    p += __shfl_xor(p, 2, 16);
    p += __shfl_xor(p, 4, 16);
    p += __shfl_xor(p, 8, 16);
    if (s == 0) {
      const float y  = p + Dd * ut;
      const float gv = g[base * CDI + d];      // bg value
      g[base * CDI + d] = y * silu_f(gv);
    }
  }
}

// ---------------------------------------------------------------------------
// 7) out-proj + bias + residual:  pre[b,l,n] = sum_e g[b,l,e]*W_out[n,e]
//                                              + b_out[n] + x[b,n,l]
// ---------------------------------------------------------------------------
__global__ __launch_bounds__(128) void k_out_proj(const float* __restrict__ g,
                                                  const float* __restrict__ W_out,
                                                  const float* __restrict__ b_out,
                                                  const float* __restrict__ x,
                                                  float* __restrict__ pre) {
  const int l0 = (blockIdx.x * 4 + threadIdx.y) * 32;
  const int n0 = blockIdx.y * 64;
  const int b  = blockIdx.z;
  const float* A = g + ((size_t)b * CL + l0) * CDI;     // row-major (L, DI)
  v8f acc[2][4] = {};
  wmma_macro_tile<2, 4>(A, CDI, 1, W_out + (size_t)n0 * CDI, CDI, CDI, acc);
  const int lane = threadIdx.x & 31, half = lane >> 4, r = lane & 15;
#pragma unroll
  for (int nj = 0; nj < 4; ++nj) {
    const int n = n0 + nj * 16 + r;
    const float bias = b_out[n];
#pragma unroll
    for (int mi = 0; mi < 2; ++mi)
#pragma unroll
      for (int i = 0; i < 8; ++i) {
        const int l = l0 + mi * 16 + 8 * half + i;
        const float v = acc[mi][nj][i] + bias + x[((size_t)b * CD + n) * CL + l];
        pre[((size_t)b * CL + l) * CD + n] = v;
      }
  }
}

// ---------------------------------------------------------------------------
// 8) LayerNorm over D per (b,l), transposed store to (B,D,L)
// ---------------------------------------------------------------------------
__global__ __launch_bounds__(256) void k_layernorm(const float* __restrict__ pre,
                                                   const float* __restrict__ ln_g,
                                                   const float* __restrict__ ln_b,
                                                   float* __restrict__ out) {
  __shared__ float s_sum[256];
  __shared__ float s_sq[256];
  const int bl  = blockIdx.x;                 // b*CL + l
  const int tid = threadIdx.x;
  const float* row = pre + (size_t)bl * CD;
  const float v0 = row[tid];
  const float v1 = row[tid + 256];
  s_sum[tid] = v0 + v1;
  s_sq[tid]  = v0 * v0 + v1 * v1;
  __syncthreads();
  for (int off = 128; off > 0; off >>= 1) {
    if (tid < off) { s_sum[tid] += s_sum[tid + off]; s_sq[tid] += s_sq[tid + off]; }
    __syncthreads();
  }
  const float mu   = s_sum[0] * (1.0f / CD);
  const float var  = s_sq[0] * (1.0f / CD) - mu * mu;
  const float rstd = rsqrtf(var + 1e-5f);
  const int b = bl / CL, l = bl % CL;
  out[((size_t)b * CD + tid) * CL + l] =
      (v0 - mu) * rstd * ln_g[tid] + ln_b[tid];
  out[((size_t)b * CD + tid + 256) * CL + l] =
      (v1 - mu) * rstd * ln_g[tid + 256] + ln_b[tid + 256];
}

// ---------------------------------------------------------------------------
extern "C" void kernel_launch(void* const* d_in, const int* in_sizes, int n_in,
                              void* d_out, int out_size, void* d_ws, size_t ws_size,
                              hipStream_t stream) {
  const float* x     = (const float*)d_in[0];
  const float* W_in  = (const float*)d_in[1];
  const float* b_in  = (const float*)d_in[2];
  const float* dw_w  = (const float*)d_in[3];
  const float* dw_b  = (const float*)d_in[4];
  const float* pw_w  = (const float*)d_in[5];
  const float* pw_b  = (const float*)d_in[6];
  const float* A_log = (const float*)d_in[7];
  const float* Dskip = (const float*)d_in[8];
  const float* W_x   = (const float*)d_in[9];
  const float* W_dt  = (const float*)d_in[10];
  const float* b_dt  = (const float*)d_in[11];
  const float* W_out = (const float*)d_in[12];
  const float* b_out = (const float*)d_in[13];
  const float* ln_g  = (const float*)d_in[14];
  const float* ln_b  = (const float*)d_in[15];
  float* out = (float*)d_out;

  float* ws = (float*)d_ws;
  const size_t BIG = (size_t)CB * CDI * CL;            // 8M floats (32 MiB)
  float* buf0 = ws;                                    // a_pre -> u -> pre
  float* buf1 = ws + BIG;                              // bg -> g
  float* buf2 = ws + 2 * BIG;                          // a_conv -> delta
  float* dtb  = ws + 3 * BIG;                          // (B,L,DT)
  float* Bmb  = dtb + (size_t)CB * CL * CDT;           // (B,L,DS)
  float* Cmb  = Bmb + (size_t)CB * CL * CDS;           // (B,L,DS)

  const dim3 gblk(32, 4, 1);                           // 4 wave32s per block
  // Each wave: 32(M)x64(N) macro-tile -> block covers 128 rows of L.
  const int GX = CL / 128;                             // 16

  // 1) in-proj (N = 2*DI = 2048)
  k_in_proj<<<dim3(GX, (2 * CDI) / 64, CB), gblk, 0, stream>>>(
      x, W_in, b_in, buf0, buf1);

  // 2) depthwise conv
  {
    const size_t total = (size_t)CB * CDI * CL;
    k_dwconv<<<dim3((unsigned)((total + 255) / 256)), dim3(256), 0, stream>>>(
        buf0, dw_w, dw_b, buf2);
  }

  // 3) pointwise conv + SiLU  (u into buf0; a_pre dead)
  k_pw_silu<<<dim3(GX, CDI / 64, CB), gblk, 0, stream>>>(
      buf2, pw_w, pw_b, buf0);

  // 4) x-proj (N = 96 -> 3 blocks of 32)
  k_xproj<<<dim3(GX, 96 / 32, CB), gblk, 0, stream>>>(
      buf0, W_x, dtb, Bmb, Cmb);

  // 5) dt-proj + softplus (delta into buf2; a_conv dead)
  k_dtproj<<<dim3(GX, CDI / 64, CB), gblk, 0, stream>>>(
      dtb, W_dt, b_dt, buf2);

  // 6) selective scan + gate (g over bg in buf1)
  k_scan_gate<<<dim3((CB * CDI * CDS) / 256), dim3(256), 0, stream>>>(
      buf2, Bmb, Cmb, buf0, A_log, Dskip, buf1);

  // 7) out-proj + residual (pre into buf0; u dead)
  k_out_proj<<<dim3(GX, CD / 64, CB), gblk, 0, stream>>>(
      buf1, W_out, b_out, x, buf0);

  // 8) LayerNorm + transpose to (B,D,L)
  k_layernorm<<<dim3(CB * CL), dim3(256), 0, stream>>>(buf0, ln_g, ln_b, out);
}